// GCNBlock_28166395527168
// MI455X (gfx1250) — compile-verified
//
#include <hip/hip_runtime.h>
#include <hip/hip_bf16.h>

typedef __attribute__((ext_vector_type(2))) float v2f;
typedef __attribute__((ext_vector_type(8))) float v8f;

#define DFEAT 128

// ---------------- zero fill ----------------
__global__ void zero_kernel(float* p, long n) {
    long i = (long)blockIdx.x * blockDim.x + threadIdx.x;
    if (i < n) p[i] = 0.0f;
}

// ---------------- degree histogram ----------------
__global__ void degree_kernel(const int* __restrict__ src,
                              const int* __restrict__ dst,
                              int* __restrict__ deg_out,
                              int* __restrict__ deg_in, int E) {
    int e = blockIdx.x * blockDim.x + threadIdx.x;
    if (e < E) {
        atomicAdd(&deg_out[src[e]], 1);
        atomicAdd(&deg_in[dst[e]], 1);
    }
}

// ---------------- deg counts -> rsqrt(max(deg,1)) in place ----------------
__global__ void norm_kernel(int* deg, int n) {
    int i = blockIdx.x * blockDim.x + threadIdx.x;
    if (i < n) {
        int d = deg[i];
        float df = (d < 1) ? 1.0f : (float)d;
        ((float*)deg)[i] = rsqrtf(df);
    }
}

// ---------------- h = (features * norm_out[:,None]) @ W via fp32 WMMA ----------------
// grid.x = N/16 row tiles; 256 threads = 8 waves; wave w computes columns [16w,16w+16)
__global__ __launch_bounds__(256) void gemm_kernel(const float* __restrict__ feat,
                                                   const float* __restrict__ W,
                                                   const float* __restrict__ norm_out,
                                                   float* __restrict__ h) {
    __shared__ float As[16 * DFEAT];   // 8 KB: scaled A tile, row-major
    const int row0 = blockIdx.x * 16;
    const int tid  = threadIdx.x;

    // cooperative load of 16x128 feature tile, pre-scaled by norm_out
    for (int i = tid; i < 16 * DFEAT; i += 256) {
        int m = i >> 7;            // 0..15
        int k = i & (DFEAT - 1);   // 0..127
        int r = row0 + m;
        As[i] = feat[(long)r * DFEAT + k] * norm_out[r];
    }
    __syncthreads();

    const int wave = tid >> 5;     // 0..7 -> output column tile
    const int lane = tid & 31;
    const int n0   = wave * 16;
    const int half = lane >> 4;    // 0: K={k,k+1}, 1: K={k+2,k+3}
    const int mn   = lane & 15;    // A: row within tile; B: column within tile

    v8f c = {};
    const float* wcol = W + n0 + mn;       // column (n0+mn) of W, stride DFEAT
    const float* arow = As + mn * DFEAT;   // row mn of scaled A tile

    #pragma unroll 4
    for (int k0 = 0; k0 < DFEAT; k0 += 4) {
        int kb = k0 + half * 2;
        v2f a = *(const v2f*)(arow + kb);          // A[mn][kb], A[mn][kb+1]
        v2f b;
        b.x = wcol[(long)kb * DFEAT];              // W[kb][n0+mn]
        b.y = wcol[(long)(kb + 1) * DFEAT];        // W[kb+1][n0+mn]
        c = __builtin_amdgcn_wmma_f32_16x16x4_f32(
                /*neg_a=*/false, a, /*neg_b=*/false, b,
                /*c_mod=*/(short)0, c, /*reuse_a=*/false, /*reuse_b=*/false);
    }

    // C layout: vgpr i, lane l -> row (l>>4)*8 + i, col (l&15)
    const int rbase = row0 + half * 8;
    #pragma unroll
    for (int i = 0; i < 8; ++i) {
        h[(long)(rbase + i) * DFEAT + n0 + mn] = c[i];
    }
}

// ---------------- scatter-add: agg[dst] += h[src], one wave per edge ----------------
__global__ __launch_bounds__(256) void scatter_kernel(const int* __restrict__ src,
                                                      const int* __restrict__ dst,
                                                      const float* __restrict__ h,
                                                      float* __restrict__ agg, int E) {
    long t = (long)blockIdx.x * blockDim.x + threadIdx.x;
    int e = (int)(t >> 5);
    int lane = (int)(t & 31);
    if (e >= E) return;
    int s = src[e];
    int d = dst[e];
    const float4* hp = (const float4*)(h + (long)s * DFEAT);
    float4 v = hp[lane];                     // 16B coalesced gather of source row
    float* op = agg + (long)d * DFEAT + lane * 4;
    atomicAdd(op + 0, v.x);
    atomicAdd(op + 1, v.y);
    atomicAdd(op + 2, v.z);
    atomicAdd(op + 3, v.w);
}

// ---------------- fused epilogue: norm_in*agg + b -> LayerNorm -> ReLU -> +features ----
// one block (128 threads = 4 waves) per node row; in-place on d_out
__global__ __launch_bounds__(128) void finalize_kernel(float* __restrict__ out,          // holds agg, rewritten
                                                       const float* __restrict__ norm_in,
                                                       const float* __restrict__ bias,
                                                       const float* __restrict__ gamma,
                                                       const float* __restrict__ beta,
                                                       const float* __restrict__ feat) {
    const int row = blockIdx.x;
    const int c   = threadIdx.x;             // 0..127
    const long idx = (long)row * DFEAT + c;

    float v = out[idx] * norm_in[row] + bias[c];

    // block-wide sum and sum-of-squares (wave32 shuffles + LDS across 4 waves)
    float s = v, ss = v * v;
    #pragma unroll
    for (int o = 16; o > 0; o >>= 1) {
        s  += __shfl_xor(s, o, 32);
        ss += __shfl_xor(ss, o, 32);
    }
    __shared__ float sbuf[4], ssbuf[4];
    if ((c & 31) == 0) { sbuf[c >> 5] = s; ssbuf[c >> 5] = ss; }
    __syncthreads();
    float ts  = sbuf[0] + sbuf[1] + sbuf[2] + sbuf[3];
    float tss = ssbuf[0] + ssbuf[1] + ssbuf[2] + ssbuf[3];

    const float inv_d = 1.0f / (float)DFEAT;
    float mu  = ts * inv_d;
    float var = tss * inv_d - mu * mu;
    float rstd = rsqrtf(var + 1e-5f);

    float n = (v - mu) * rstd * gamma[c] + beta[c];
    float r = n > 0.0f ? n : 0.0f;
    out[idx] = r + feat[idx];
}

extern "C" void kernel_launch(void* const* d_in, const int* in_sizes, int n_in,
                              void* d_out, int out_size, void* d_ws, size_t ws_size,
                              hipStream_t stream) {
    const float* feat     = (const float*)d_in[0];
    const float* W        = (const float*)d_in[1];
    const float* bias     = (const float*)d_in[2];
    const float* gamma    = (const float*)d_in[3];
    const float* beta     = (const float*)d_in[4];
    const int*   edge_src = (const int*)d_in[5];
    const int*   edge_dst = (const int*)d_in[6];

    const int N = in_sizes[0] / DFEAT;       // 50000
    const int E = in_sizes[5];               // 800000

    // workspace layout: [deg_out:N ints][deg_in:N ints][pad][h: N*DFEAT floats]
    int*   deg_out  = (int*)d_ws;
    int*   deg_in   = deg_out + N;
    float* norm_out = (float*)deg_out;       // overwritten in place by norm_kernel
    float* norm_in  = (float*)deg_in;
    long   hoff     = ((long)(2 * N) + 127) & ~127L;   // float-element offset, 512B aligned
    float* h        = (float*)d_ws + hoff;

    float* out = (float*)d_out;              // doubles as agg accumulator

    // 1) zero degree counters
    {
        long n = 2L * N;
        zero_kernel<<<(int)((n + 255) / 256), 256, 0, stream>>>((float*)d_ws, n);
    }
    // 2) zero aggregation buffer (d_out)
    {
        long n = (long)N * DFEAT;
        zero_kernel<<<(int)((n + 255) / 256), 256, 0, stream>>>(out, n);
    }
    // 3) degree histogram
    degree_kernel<<<(E + 255) / 256, 256, 0, stream>>>(edge_src, edge_dst, deg_out, deg_in, E);
    // 4) counts -> rsqrt(max(deg,1)) in place (both arrays contiguous)
    norm_kernel<<<(2 * N + 255) / 256, 256, 0, stream>>>(deg_out, 2 * N);
    // 5) h = (feat * norm_out) @ W  via fp32 WMMA
    gemm_kernel<<<N / 16, 256, 0, stream>>>(feat, W, norm_out, h);
    // 6) scatter-add over edges: one wave per edge
    {
        long threads = (long)E * 32;
        scatter_kernel<<<(int)((threads + 255) / 256), 256, 0, stream>>>(edge_src, edge_dst, h, out, E);
    }
    // 7) fused normalization + LayerNorm + ReLU + residual, in place
    finalize_kernel<<<N, DFEAT, 0, stream>>>(out, norm_in, bias, gamma, beta, feat);
}